// EPAll2AllLayer_54563264529075
// MI455X (gfx1250) — compile-verified
//
#include <hip/hip_runtime.h>
#include <hip/hip_bf16.h>
#include <stdint.h>

// Problem constants (from the reference)
#define TOKENS      8192
#define HIDDEN      4096
#define TOPK        2
#define NEXP        64
#define NFLAT       (TOKENS * TOPK)   // 16384 flat (token,k) slots
#define CHUNK       256               // slots per chunk for the stable counting sort
#define NCHUNK      (NFLAT / CHUNK)   // 64 chunks

// ---------------------------------------------------------------------------
// Kernel 1: per-chunk expert histogram.  64 blocks x 256 threads.
// hist[c*NEXP + e] = #slots in chunk c assigned to expert e.
// ---------------------------------------------------------------------------
__global__ void k_hist(const int* __restrict__ eidx, int* __restrict__ hist) {
    __shared__ int h[NEXP];
    const int t = threadIdx.x;
    if (t < NEXP) h[t] = 0;
    __syncthreads();
    const int slot = blockIdx.x * CHUNK + t;
    const int e = eidx[slot];
    atomicAdd(&h[e], 1);
    __syncthreads();
    if (t < NEXP) hist[blockIdx.x * NEXP + t] = h[t];
}

// ---------------------------------------------------------------------------
// Kernel 2: scans.  1 block x 64 threads.
//   chunkbase[c*NEXP+e] = # of expert-e slots in chunks < c   (exclusive)
//   offs[e]             = exclusive cumsum of per-expert totals
// ---------------------------------------------------------------------------
__global__ void k_scan(const int* __restrict__ hist,
                       int* __restrict__ chunkbase,
                       int* __restrict__ offs) {
    __shared__ int tot[NEXP];
    const int e = threadIdx.x;   // 0..63
    int run = 0;
    for (int c = 0; c < NCHUNK; ++c) {
        chunkbase[c * NEXP + e] = run;
        run += hist[c * NEXP + e];
    }
    tot[e] = run;
    __syncthreads();
    if (e == 0) {
        int acc = 0;
        for (int i = 0; i < NEXP; ++i) { offs[i] = acc; acc += tot[i]; }
    }
}

// ---------------------------------------------------------------------------
// Kernel 3: stable intra-chunk rank -> gather table.  64 blocks x 256 threads.
//   out_row for slot i = offs[e] + chunkbase[chunk][e] + (#earlier same-e in chunk)
//   gather[out_row] = i >> 1  (source token, TOPK == 2)
// ---------------------------------------------------------------------------
__global__ void k_rank(const int* __restrict__ eidx,
                       const int* __restrict__ chunkbase,
                       const int* __restrict__ offs,
                       int* __restrict__ gather) {
    __shared__ int se[CHUNK];
    const int t = threadIdx.x;
    const int b = blockIdx.x;
    const int slot = b * CHUNK + t;
    const int e = eidx[slot];
    se[t] = e;
    __syncthreads();
    int rank = 0;
    for (int j = 0; j < t; ++j) rank += (se[j] == e) ? 1 : 0;
    const int dest = offs[e] + chunkbase[b * NEXP + e] + rank;
    gather[dest] = slot >> 1;   // order // topk
}

// ---------------------------------------------------------------------------
// CDNA5 async global<->LDS DMA helpers (ASYNCcnt-tracked, no VGPR data path).
// LDS byte address = low 32 bits of the generic shared pointer (aperture bits
// occupy [63:32] per the ISA aperture check).
//
// Cache policy: loads stay RT (input rows are re-read once; 128 MB input fits
// the 192 MB L2).  Stores are NT so the 256 MB write-once output stream does
// not evict the input from L2 before its second gather.
// ---------------------------------------------------------------------------
__device__ __forceinline__ void async_load_b128_to_lds(unsigned lds_off, const void* gsrc) {
    asm volatile("global_load_async_to_lds_b128 %0, %1, off"
                 :: "v"(lds_off), "v"(gsrc)
                 : "memory");
}
__device__ __forceinline__ void async_store_b128_from_lds_nt(void* gdst, unsigned lds_off) {
    asm volatile("global_store_async_from_lds_b128 %0, %1, off th:TH_STORE_NT"
                 :: "v"(gdst), "v"(lds_off)
                 : "memory");
}
__device__ __forceinline__ void wait_async_0() {
#if __has_builtin(__builtin_amdgcn_s_wait_asynccnt)
    __builtin_amdgcn_s_wait_asynccnt(0);
#else
    asm volatile("s_wait_asynccnt 0" ::: "memory");
#endif
}

// ---------------------------------------------------------------------------
// Kernel 4: the big copy.  One block per output row (16384 blocks x 256 thr).
// Row = 16 KB.  Each thread DMAs 4 x 16B chunks global->LDS, waits ASYNCcnt,
// then DMAs the same LDS bytes LDS->global.  Each wave only touches LDS bytes
// its own lanes loaded, so no __syncthreads is required; the final ASYNCcnt
// wait (and S_ENDPGM's implicit wait-idle) drains the stores.
// ---------------------------------------------------------------------------
__global__ void k_copy(const float* __restrict__ x,
                       const int* __restrict__ gather,
                       float* __restrict__ out) {
    __shared__ float smem[HIDDEN];           // 16 KB staging
    const int row = blockIdx.x;
    const int src = gather[row];             // uniform across the block
    const float* gsrc = x   + (size_t)src * HIDDEN;
    float*       gdst = out + (size_t)row * HIDDEN;

    const unsigned lbase = (unsigned)(size_t)(const void*)smem;  // LDS byte offset
    const int t = threadIdx.x;               // 0..255

    // 256 threads x 16B = 4 KB per pass; 4 passes per 16 KB row.
#pragma unroll
    for (int k = 0; k < 4; ++k) {
        const unsigned loff = lbase + (unsigned)(t * 16 + k * 4096);
        async_load_b128_to_lds(loff, (const void*)(gsrc + t * 4 + k * 1024));
    }
    wait_async_0();                          // this wave's loads have landed in LDS
#pragma unroll
    for (int k = 0; k < 4; ++k) {
        const unsigned loff = lbase + (unsigned)(t * 16 + k * 4096);
        async_store_b128_from_lds_nt((void*)(gdst + t * 4 + k * 1024), loff);
    }
    wait_async_0();                          // explicit drain (S_ENDPGM also waits)
}

// ---------------------------------------------------------------------------
// Launch
// ---------------------------------------------------------------------------
extern "C" void kernel_launch(void* const* d_in, const int* in_sizes, int n_in,
                              void* d_out, int out_size, void* d_ws, size_t ws_size,
                              hipStream_t stream) {
    const float* x    = (const float*)d_in[0];   // [TOKENS, HIDDEN] f32
    const int*   eidx = (const int*)  d_in[1];   // [TOKENS, TOPK]   i32
    float*       out  = (float*)d_out;           // [NFLAT, HIDDEN]  f32

    int* ws        = (int*)d_ws;
    int* hist      = ws;                 // NCHUNK*NEXP = 4096 ints
    int* chunkbase = ws + 4096;          // NCHUNK*NEXP = 4096 ints
    int* offs      = ws + 8192;          // NEXP = 64 ints
    int* gather    = ws + 8256;          // NFLAT = 16384 ints  (~96 KB total)

    k_hist<<<NCHUNK, CHUNK, 0, stream>>>(eidx, hist);
    k_scan<<<1, NEXP, 0, stream>>>(hist, chunkbase, offs);
    k_rank<<<NCHUNK, CHUNK, 0, stream>>>(eidx, chunkbase, offs, gather);
    k_copy<<<NFLAT, 256, 0, stream>>>(x, gather, out);
}